// CostVolume_72267119722848
// MI455X (gfx1250) — compile-verified
//
#include <hip/hip_runtime.h>
#include <hip/hip_bf16.h>
#include <stdint.h>

// Cost volume: cost[b,c,h,w,d] = left[b,c,h,w] - right[b,c,h,w-d], 0 if w<d.
// B=4 C=32 H=128 W=240 D=24. Pure store-BW bound (~409MB @ 23.3TB/s ~ 17.6us).
// CDNA5 paths used: global_load_async_to_lds_b128 (ASYNCcnt) for input staging,
// s_wait_asynccnt, and non-temporal b128 streaming stores for the 377MB output.

typedef float v4f __attribute__((ext_vector_type(4)));

#define W_IMG 240
#define D_LVL 24
#define ROWS  4      // rows of W per block (contiguous in memory)
#define TPB   256    // 8 wave32

__device__ __forceinline__ uint32_t lds_off(const void* p) {
    // Low 32 bits of a generic shared pointer == wave-relative LDS byte offset
    return (uint32_t)(uintptr_t)p;
}

__device__ __forceinline__ void async_b128_to_lds(const float* gaddr, uint32_t loff) {
    asm volatile("global_load_async_to_lds_b128 %0, %1, off"
                 :: "v"(loff), "v"(gaddr)
                 : "memory");
}

__global__ __launch_bounds__(TPB)
void cost_volume_kernel(const float* __restrict__ left,
                        const float* __restrict__ right,
                        float* __restrict__ out)
{
    __shared__ float sL[ROWS * W_IMG];   // 3840 B
    __shared__ float sR[ROWS * W_IMG];   // 3840 B

    const int t   = threadIdx.x;
    const int blk = blockIdx.x;

    const float* lBase = left  + (size_t)blk * (ROWS * W_IMG);
    const float* rBase = right + (size_t)blk * (ROWS * W_IMG);

    // ---- Stage 1: async global -> LDS (ASYNCcnt path) -------------------
    // 480 x 16B chunks total: 240 for left, 240 for right.
    constexpr int CHUNKS_PER_IMG = ROWS * W_IMG / 4;   // 240
    constexpr int CHUNKS_TOTAL   = 2 * CHUNKS_PER_IMG; // 480
    #pragma unroll
    for (int pass = 0; pass < 2; ++pass) {
        int i = t + pass * TPB;
        if (i < CHUNKS_TOTAL) {
            if (i < CHUNKS_PER_IMG) {
                async_b128_to_lds(lBase + i * 4, lds_off(&sL[i * 4]));
            } else {
                int j = i - CHUNKS_PER_IMG;
                async_b128_to_lds(rBase + j * 4, lds_off(&sR[j * 4]));
            }
        }
    }
    asm volatile("s_wait_asynccnt 0" ::: "memory");
    __syncthreads();

    // ---- Stage 2: compute + coalesced non-temporal b128 stores ----------
    float* oBase = out + (size_t)blk * (ROWS * W_IMG * D_LVL);
    constexpr int QPR    = W_IMG * D_LVL / 4;   // 1440 float4 per row
    constexpr int QPW    = D_LVL / 4;           // 6 float4 per (r,w)
    constexpr int NQUADS = ROWS * QPR;          // 5760 float4 per block
    constexpr int NITER  = (NQUADS + TPB - 1) / TPB;  // 23

    for (int k = 0; k < NITER; ++k) {
        int id = t + k * TPB;                 // contiguous float4 index -> coalesced
        if (id < NQUADS) {
            int r   = id / QPR;
            int rem = id - r * QPR;
            int w   = rem / QPW;
            int dv  = rem - w * QPW;

            float l = sL[r * W_IMG + w];
            v4f o;
            #pragma unroll
            for (int j = 0; j < 4; ++j) {
                int d   = dv * 4 + j;
                int idx = w - d;
                float rv = sR[r * W_IMG + (idx < 0 ? 0 : idx)];
                o[j] = (idx >= 0) ? (l - rv) : 0.0f;
            }
            __builtin_nontemporal_store(o, (v4f*)(oBase + (size_t)id * 4));
        }
    }
}

extern "C" void kernel_launch(void* const* d_in, const int* in_sizes, int n_in,
                              void* d_out, int out_size, void* d_ws, size_t ws_size,
                              hipStream_t stream) {
    const float* left  = (const float*)d_in[0];
    const float* right = (const float*)d_in[1];
    float*       out   = (float*)d_out;

    // B*C*H rows of W floats; 4 rows per block -> 4096 blocks for this shape.
    int nRows  = in_sizes[0] / W_IMG;
    int blocks = nRows / ROWS;

    cost_volume_kernel<<<blocks, TPB, 0, stream>>>(left, right, out);
}